// SubgraphingLayer_42502996361386
// MI455X (gfx1250) — compile-verified
//
#include <hip/hip_runtime.h>
#include <hip/hip_bf16.h>

// Problem constants (from reference setup_inputs): B=8, N=2048, F=64, K=32
#define BB 8
#define NN 2048
#define FF 64
#define KK 32

// Native clang vector type (NOT HIP_vector_type) so nontemporal builtins accept it.
typedef float vfloat4 __attribute__((ext_vector_type(4)));

// ---------------------------------------------------------------------------
// Kernel 1: wfeat[b,n,k,0:64] = features[b, neigh[n,k], 0:64]
// One thread per 128-bit chunk of output. 16 consecutive threads copy one
// 256-byte feature row -> fully coalesced global_load_b128 / global_store_b128.
// features is 4MB total and heavily reused -> keep cached (RT loads);
// output is written once and never re-read -> non-temporal store.
// ---------------------------------------------------------------------------
__global__ void __launch_bounds__(256)
wfeat_gather_kernel(const float* __restrict__ features,
                    const int*   __restrict__ neigh,
                    float*       __restrict__ wfeat) {
    // total vfloat4 elements = B*N*K*(F/4) = 8*2048*32*16 = 8388608
    unsigned int tid = blockIdx.x * blockDim.x + threadIdx.x;

    unsigned int f4 = tid & 15u;          // 128b chunk within row (0..15)
    unsigned int k  = (tid >> 4) & 31u;   // neighbor slot (0..31)
    unsigned int n  = (tid >> 9) & 2047u; // node (0..2047)
    unsigned int b  = tid >> 20;          // batch (0..7)

    int r = neigh[n * KK + k];            // broadcast across the 16-lane group

    const vfloat4* src =
        reinterpret_cast<const vfloat4*>(features + ((long)b * NN + r) * FF) + f4;
    vfloat4 v = *src;

    vfloat4* dst = reinterpret_cast<vfloat4*>(wfeat) + tid;
    __builtin_nontemporal_store(v, dst);  // NT: stream 128MB past L2
}

// ---------------------------------------------------------------------------
// Kernel 2: wadj[b,n,i,j]   = adj[b,   neigh[n,i], neigh[n,j]]
//           wedges[b,n,i,j] = edges[b, neigh[n,i], neigh[n,j]]
// One wave32 per (b,n) pair. Lane j holds its column index c_j; the row index
// r_i is the SAME index vector, broadcast from lane i via wave shuffle.
// Per iteration: 2 scattered 4B gathers (adj+edges share the address) and
// 2 perfectly coalesced 128B row stores (non-temporal).
// A gfx1250 global_prefetch_b8 pulls next row's lines one iteration early.
// ---------------------------------------------------------------------------
__global__ void __launch_bounds__(256)
wadj_gather_kernel(const float* __restrict__ adj,
                   const float* __restrict__ edges,
                   const int*   __restrict__ neigh,
                   float*       __restrict__ wadj,
                   float*       __restrict__ wedges) {
    unsigned int flat = blockIdx.x * blockDim.x + threadIdx.x;
    unsigned int wave = flat >> 5;        // 0 .. B*N-1  (16384 waves)
    unsigned int lane = threadIdx.x & 31u;

    unsigned int b = wave / NN;
    unsigned int n = wave % NN;

    int cj = neigh[n * KK + lane];        // this lane's column index

    const long  base  = (long)b * NN * NN;           // adj/edges batch base
    const long  obase = ((long)(b * NN + n)) * (KK * KK);

    // Prime: prefetch row 0's gather lines for this lane.
    {
        int r0 = __shfl(cj, 0, 32);
        __builtin_prefetch(adj   + base + (long)r0 * NN + cj, 0, 0);
        __builtin_prefetch(edges + base + (long)r0 * NN + cj, 0, 0);
    }

    for (int i = 0; i < KK; ++i) {
        int r = __shfl(cj, i, 32);        // row index == shared index vector
        long off = base + (long)r * NN + cj;

        // Prefetch next row's scattered lines (gfx1250 global_prefetch_b8).
        if (i + 1 < KK) {
            int rn = __shfl(cj, i + 1, 32);
            long noff = base + (long)rn * NN + cj;
            __builtin_prefetch(adj   + noff, 0, 0);
            __builtin_prefetch(edges + noff, 0, 0);
        }

        float a = adj[off];               // scattered 4B gather (cached, RT)
        float e = edges[off];

        long o = obase + (long)i * KK + lane;   // coalesced 128B row store
        __builtin_nontemporal_store(a, wadj   + o);
        __builtin_nontemporal_store(e, wedges + o);
    }
}

// ---------------------------------------------------------------------------
// Host-side launcher
// ---------------------------------------------------------------------------
extern "C" void kernel_launch(void* const* d_in, const int* in_sizes, int n_in,
                              void* d_out, int out_size, void* d_ws, size_t ws_size,
                              hipStream_t stream) {
    const float* features = (const float*)d_in[0];   // (8,2048,64)   f32
    const float* adj      = (const float*)d_in[1];   // (8,2048,2048) f32
    const float* edges    = (const float*)d_in[2];   // (8,2048,2048) f32
    const int*   neigh    = (const int*)  d_in[3];   // (2048,32)     i32

    float* out    = (float*)d_out;
    float* wfeat  = out;                                        // 33554432 floats
    float* wadj   = out + (long)BB * NN * KK * FF;              // 16777216 floats
    float* wedges = wadj + (long)BB * NN * KK * KK;             // 16777216 floats

    // Kernel 1: 8388608 128-bit copies, 256 threads/block
    {
        const unsigned int total = BB * NN * KK * (FF / 4);     // 8388608
        wfeat_gather_kernel<<<total / 256, 256, 0, stream>>>(features, neigh, wfeat);
    }

    // Kernel 2: one wave32 per (b,n): 16384 waves -> 2048 blocks of 256
    {
        const unsigned int waves = BB * NN;                     // 16384
        wadj_gather_kernel<<<(waves * 32) / 256, 256, 0, stream>>>(
            adj, edges, neigh, wadj, wedges);
    }
}